// Generator_54391465836718
// MI455X (gfx1250) — compile-verified
//
#include <hip/hip_runtime.h>

typedef __attribute__((ext_vector_type(2))) float v2f;
typedef __attribute__((ext_vector_type(8))) float v8f;

#define S_ 30
#define B_ 16
#define L_ 256
#define H_ 400
#define V_ 20000
#define T_ 8
#define G_ 3
#define BB_ (S_ * B_)

// ---------------------------------------------------------------------------
// Optional Tensor Data Mover support (gfx1250). Guarded so the file compiles
// on toolchains without the builtin; arity differs between ROCm 7.2 (5 args)
// and amdgpu-toolchain/therock (6 args) -- detected via the TDM header.
// ---------------------------------------------------------------------------
#if __has_builtin(__builtin_amdgcn_tensor_load_to_lds)
#define HAVE_TDM 1
typedef __attribute__((ext_vector_type(4))) unsigned int u32x4_t;
typedef __attribute__((ext_vector_type(8))) int i32x8_t;
typedef __attribute__((ext_vector_type(4))) int i32x4_t;

// 1-row 2D tile copy: nfloats f32 from global -> LDS (D# per ISA ch.8).
__device__ __forceinline__ void tdm_load_row_to_lds(unsigned lds_addr,
                                                    const void* gptr,
                                                    unsigned nfloats) {
  unsigned long long ga = (unsigned long long)(size_t)gptr;
  u32x4_t g0;
  g0[0] = 1u;                                               // count=1 (valid)
  g0[1] = lds_addr;                                         // LDS byte addr
  g0[2] = (unsigned)(ga & 0xffffffffull);                   // global_addr lo
  g0[3] = (unsigned)((ga >> 32) & 0x01ffffffull) | (2u << 30); // hi | type=2
  i32x8_t g1;
  g1[0] = (int)(2u << 16);            // data_size = 4 bytes
  g1[1] = (int)(nfloats << 16);       // tensor_dim0[15:0]
  g1[2] = (int)(1u << 16);            // tensor_dim1 = 1
  g1[3] = (int)(nfloats << 16);       // tile_dim0
  g1[4] = 1;                          // tile_dim1 = 1
  g1[5] = (int)nfloats;               // tensor_dim0_stride
  g1[6] = 0;
  g1[7] = 0;
  i32x4_t gz = {0, 0, 0, 0};
#if __has_include(<hip/amd_detail/amd_gfx1250_TDM.h>)
  i32x8_t gz8 = {0, 0, 0, 0, 0, 0, 0, 0};
  __builtin_amdgcn_tensor_load_to_lds(g0, g1, gz, gz, gz8, 0);
#else
  __builtin_amdgcn_tensor_load_to_lds(g0, g1, gz, gz, 0);
#endif
}
#endif

// ---------------------------------------------------------------------------
// fp32 WMMA GEMM:  C[M,N] = A[M,K] * B[N,K]^T   (all row-major)
// V_WMMA_F32_16X16X4_F32; wave tile = 32x80 (2 M-tiles x 5 N-tiles).
// Register double-buffered: loads for k+4 are issued before the WMMAs that
// consume the k fragments, so waits are non-blocking (loadcnt<=7, not 0).
// Requirements: M%32==0, N%80==0, K%8==0. Fragment layout (ISA 7.12.2):
// A/B element [r=lane&15][k=2*(lane>>4)+v]; D [m=j+8*(lane>>4)][n=lane&15].
// ---------------------------------------------------------------------------
__global__ __launch_bounds__(256) void wmma_gemm_nt(
    const float* __restrict__ A, const float* __restrict__ Bm,
    float* __restrict__ C, int M, int N, int K) {
  constexpr int NTI = 5;
  const int gwave = (blockIdx.x * blockDim.x + threadIdx.x) >> 5;
  const int lane  = threadIdx.x & 31;
  const int ngroups = N / (16 * NTI);
  const int total   = (M / 32) * ngroups;
  if (gwave >= total) return;                 // wave-uniform: EXEC stays full
  const int mg = gwave / ngroups;
  const int ng = gwave % ngroups;
  const int m0 = mg * 32;
  const int n0 = ng * 16 * NTI;
  const int r  = lane & 15;
  const int kh = (lane >> 4) << 1;            // 0 or 2

  const float* ap0 = A + (size_t)(m0 + r) * K + kh;
  const float* ap1 = ap0 + (size_t)16 * K;
  const float* bp[NTI];
#pragma unroll
  for (int i = 0; i < NTI; i++) bp[i] = Bm + (size_t)(n0 + i * 16 + r) * K + kh;

  v8f acc[2][NTI] = {};
  v2f a_c[2], b_c[NTI];
  a_c[0] = *(const v2f*)ap0;
  a_c[1] = *(const v2f*)ap1;
#pragma unroll
  for (int i = 0; i < NTI; i++) b_c[i] = *(const v2f*)bp[i];

  for (int k0 = 4; k0 < K; k0 += 4) {
    v2f a_n[2], b_n[NTI];
    a_n[0] = *(const v2f*)(ap0 + k0);         // issue next-step loads first
    a_n[1] = *(const v2f*)(ap1 + k0);
#pragma unroll
    for (int i = 0; i < NTI; i++) b_n[i] = *(const v2f*)(bp[i] + k0);
#pragma unroll
    for (int m = 0; m < 2; m++)
#pragma unroll
      for (int i = 0; i < NTI; i++)
        acc[m][i] = __builtin_amdgcn_wmma_f32_16x16x4_f32(
            false, a_c[m], false, b_c[i], (short)0, acc[m][i], false, false);
    a_c[0] = a_n[0]; a_c[1] = a_n[1];
#pragma unroll
    for (int i = 0; i < NTI; i++) b_c[i] = b_n[i];
  }
#pragma unroll
  for (int m = 0; m < 2; m++)
#pragma unroll
    for (int i = 0; i < NTI; i++)
      acc[m][i] = __builtin_amdgcn_wmma_f32_16x16x4_f32(
          false, a_c[m], false, b_c[i], (short)0, acc[m][i], false, false);

  const int moff = (lane >> 4) * 8;
#pragma unroll
  for (int m = 0; m < 2; m++)
#pragma unroll
    for (int i = 0; i < NTI; i++)
#pragma unroll
      for (int j = 0; j < 8; j++)
        C[(size_t)(m0 + m * 16 + moff + j) * N + n0 + i * 16 + r] = acc[m][i][j];
}

// ---------------------------------------------------------------------------
// x0 = slot_emb[domain] + slot_emb[slot]; h0 = tiled encoded_hidden.
// ---------------------------------------------------------------------------
__global__ void init_state_kernel(const float* __restrict__ enc_hidden,
                                  const float* __restrict__ slot_emb,
                                  const int* __restrict__ domain_idx,
                                  const int* __restrict__ slot_idx,
                                  float* __restrict__ x, float* __restrict__ h) {
  int idx = blockIdx.x * blockDim.x + threadIdx.x;
  if (idx >= BB_ * H_) return;
  int bb = idx / H_, k = idx % H_;
  int s = bb / B_, b = bb % B_;
  x[idx] = slot_emb[domain_idx[s] * H_ + k] + slot_emb[slot_idx[s] * H_ + k];
  h[idx] = enc_hidden[b * H_ + k];
}

// ---------------------------------------------------------------------------
// GRU gate fusion from pre-activations gi = x@Wih^T, gh = h@Whh^T.
// ---------------------------------------------------------------------------
__global__ void gru_fuse_kernel(const float* __restrict__ gi, const float* __restrict__ gh,
                                const float* __restrict__ b_ih, const float* __restrict__ b_hh,
                                const float* __restrict__ h_old, float* __restrict__ h_new) {
  int idx = blockIdx.x * blockDim.x + threadIdx.x;
  if (idx >= BB_ * H_) return;
  int bb = idx / H_, j = idx % H_;
  const float* gir = gi + (size_t)bb * 3 * H_;
  const float* ghr = gh + (size_t)bb * 3 * H_;
  float ir = gir[j]          + b_ih[j],          hr = ghr[j]          + b_hh[j];
  float iz = gir[H_ + j]     + b_ih[H_ + j],     hz = ghr[H_ + j]     + b_hh[H_ + j];
  float in = gir[2 * H_ + j] + b_ih[2 * H_ + j], hn = ghr[2 * H_ + j] + b_hh[2 * H_ + j];
  float rg = 1.f / (1.f + expf(-(ir + hr)));
  float zg = 1.f / (1.f + expf(-(iz + hz)));
  float ng = tanhf(in + rg * hn);
  h_new[idx] = (1.f - zg) * ng + zg * h_old[idx];
}

// ---------------------------------------------------------------------------
// Masked attention. The 1.6KB h_new row broadcast into LDS is staged by the
// Tensor Data Mover (one wave issues tensor_load_to_lds + s_wait_tensorcnt,
// the workgroup barrier publishes it). Fallback: plain LDS fill.
// ---------------------------------------------------------------------------
__global__ __launch_bounds__(256) void attention_kernel(
    const float* __restrict__ hnew, const float* __restrict__ enc,
    const int* __restrict__ lens, float* __restrict__ prob, float* __restrict__ ctx) {
  int bb = blockIdx.x;
  int b  = bb % B_;
  int tid = threadIdx.x;
  __shared__ float sh_h[H_];
  __shared__ float sh_p[L_];
  __shared__ float red[256];

#ifdef HAVE_TDM
  if (tid < 32) {
    tdm_load_row_to_lds((unsigned)(size_t)(void*)sh_h,
                        hnew + (size_t)bb * H_, H_);
    __builtin_amdgcn_s_wait_tensorcnt(0);
  }
  __syncthreads();
#else
  for (int k = tid; k < H_; k += 256) sh_h[k] = hnew[(size_t)bb * H_ + k];
  __syncthreads();
#endif

  int len = lens[b];
  const float* erow = enc + ((size_t)b * L_ + tid) * H_;
  float sc = 0.f;
  for (int k = 0; k < H_; k++) sc += sh_h[k] * erow[k];
  float val = (tid < len) ? sc : -__builtin_inff();

  red[tid] = val; __syncthreads();
  for (int st = 128; st > 0; st >>= 1) {
    if (tid < st) red[tid] = fmaxf(red[tid], red[tid + st]);
    __syncthreads();
  }
  float mx = red[0]; __syncthreads();

  float e = (tid < len) ? expf(sc - mx) : 0.f;
  red[tid] = e; __syncthreads();
  for (int st = 128; st > 0; st >>= 1) {
    if (tid < st) red[tid] += red[tid + st];
    __syncthreads();
  }
  float p = e / red[0];
  sh_p[tid] = p;
  prob[(size_t)bb * L_ + tid] = p;
  __syncthreads();

  for (int k = tid; k < H_; k += 256) {
    float c = 0.f;
    for (int l = 0; l < L_; l++) c += sh_p[l] * enc[((size_t)b * L_ + l) * H_ + k];
    ctx[(size_t)bb * H_ + k] = c;
  }
}

// ---------------------------------------------------------------------------
// Pointer/generate switch: sigmoid([h|ctx|x] . w_ratio + b). One wave per row.
// ---------------------------------------------------------------------------
__global__ void switch_kernel(const float* __restrict__ hnew, const float* __restrict__ ctx,
                              const float* __restrict__ x, const float* __restrict__ wr,
                              const float* __restrict__ wrb, float* __restrict__ sw) {
  int wid  = (blockIdx.x * blockDim.x + threadIdx.x) >> 5;
  int lane = threadIdx.x & 31;
  if (wid >= BB_) return;
  float acc = 0.f;
  for (int i = lane; i < 3 * H_; i += 32) {
    float v;
    if (i < H_)           v = hnew[(size_t)wid * H_ + i];
    else if (i < 2 * H_)  v = ctx[(size_t)wid * H_ + i - H_];
    else                  v = x[(size_t)wid * H_ + i - 2 * H_];
    acc += v * wr[i];
  }
  for (int off = 16; off > 0; off >>= 1) acc += __shfl_down(acc, off);
  if (lane == 0) sw[wid] = 1.f / (1.f + expf(-(acc + wrb[0])));
}

// ---------------------------------------------------------------------------
// Gates from the t==0 context: out[b,s,g] = ctx[bb] . w_gate[g] + b[g].
// ---------------------------------------------------------------------------
__global__ void gate_kernel(const float* __restrict__ ctx, const float* __restrict__ wg,
                            const float* __restrict__ wgb, float* __restrict__ out_gate) {
  int idx = blockIdx.x * blockDim.x + threadIdx.x;
  if (idx >= BB_ * G_) return;
  int bb = idx / G_, g = idx % G_;
  int s = bb / B_, b = bb % B_;
  float acc = wgb[g];
  const float* c = ctx + (size_t)bb * H_;
  const float* w = wg + (size_t)g * H_;
  for (int k = 0; k < H_; k++) acc += c[k] * w[k];
  out_gate[((size_t)b * S_ + s) * G_ + g] = acc;
}

// ---------------------------------------------------------------------------
// Per-row softmax over V; writes switch * p_vocab to the output slice.
// ---------------------------------------------------------------------------
__global__ __launch_bounds__(256) void vocab_softmax_write_kernel(
    const float* __restrict__ logits, const float* __restrict__ sw,
    float* __restrict__ out, int t) {
  int bb = blockIdx.x;
  int tid = threadIdx.x;
  __shared__ float red[256];
  const float* row = logits + (size_t)bb * V_;

  float mx = -__builtin_inff();
  for (int v = tid; v < V_; v += 256) {
    if (v + 4096 < V_) __builtin_prefetch(row + v + 4096, 0, 0);
    mx = fmaxf(mx, row[v]);
  }
  red[tid] = mx; __syncthreads();
  for (int st = 128; st > 0; st >>= 1) {
    if (tid < st) red[tid] = fmaxf(red[tid], red[tid + st]);
    __syncthreads();
  }
  mx = red[0]; __syncthreads();

  float s = 0.f;
  for (int v = tid; v < V_; v += 256) s += expf(row[v] - mx);
  red[tid] = s; __syncthreads();
  for (int st = 128; st > 0; st >>= 1) {
    if (tid < st) red[tid] += red[tid + st];
    __syncthreads();
  }
  float scale = sw[bb] / red[0];

  int sl = bb / B_, b = bb % B_;
  float* orow = out + (((size_t)b * S_ + sl) * T_ + t) * V_;
  for (int v = tid; v < V_; v += 256) orow[v] = scale * expf(row[v] - mx);
}

// ---------------------------------------------------------------------------
// Copy distribution: atomically add (1-switch)*prob[bb,l] onto out[.., story].
// ---------------------------------------------------------------------------
__global__ void scatter_copy_kernel(const float* __restrict__ prob, const float* __restrict__ sw,
                                    const int* __restrict__ story, const int* __restrict__ lens,
                                    float* __restrict__ out, int t) {
  int idx = blockIdx.x * blockDim.x + threadIdx.x;
  if (idx >= BB_ * L_) return;
  int bb = idx / L_, l = idx % L_;
  int sl = bb / B_, b = bb % B_;
  if (l >= lens[b]) return;
  float w = (1.f - sw[bb]) * prob[(size_t)bb * L_ + l];
  int v = story[b * L_ + l];
  atomicAdd(out + (((size_t)b * S_ + sl) * T_ + t) * V_ + v, w);
}

// ---------------------------------------------------------------------------
// Teacher forcing: x_next = emb_weight[target_batches[b,s,t]].
// ---------------------------------------------------------------------------
__global__ void teacher_gather_kernel(const float* __restrict__ emb,
                                      const int* __restrict__ targets,
                                      float* __restrict__ x, int t) {
  int idx = blockIdx.x * blockDim.x + threadIdx.x;
  if (idx >= BB_ * H_) return;
  int bb = idx / H_, k = idx % H_;
  int s = bb / B_, b = bb % B_;
  int wi = targets[((size_t)b * S_ + s) * T_ + t];
  x[idx] = emb[(size_t)wi * H_ + k];
}

// ---------------------------------------------------------------------------
extern "C" void kernel_launch(void* const* d_in, const int* in_sizes, int n_in,
                              void* d_out, int out_size, void* d_ws, size_t ws_size,
                              hipStream_t stream) {
  (void)in_sizes; (void)n_in; (void)out_size; (void)ws_size;

  const float* enc_hidden = (const float*)d_in[0];   // [1,B,H]
  const float* enc_out    = (const float*)d_in[1];   // [B,L,H]
  const float* emb        = (const float*)d_in[2];   // [V,H]
  const float* w_ih       = (const float*)d_in[3];   // [3H,H]
  const float* w_hh       = (const float*)d_in[4];   // [3H,H]
  const float* b_ih       = (const float*)d_in[5];   // [3H]
  const float* b_hh       = (const float*)d_in[6];   // [3H]
  const float* wr         = (const float*)d_in[7];   // [1,3H]
  const float* wrb        = (const float*)d_in[8];   // [1]
  const float* wg         = (const float*)d_in[9];   // [G,H]
  const float* wgb        = (const float*)d_in[10];  // [G]
  const float* slot_emb   = (const float*)d_in[11];  // [NS,H]
  const int*   lens       = (const int*)d_in[12];    // [B]
  const int*   story      = (const int*)d_in[13];    // [B,L]
  const int*   targets    = (const int*)d_in[14];    // [B,S,T]
  const int*   domain_idx = (const int*)d_in[15];    // [S]
  const int*   slot_idx   = (const int*)d_in[16];    // [S]

  float* out      = (float*)d_out;                              // [B,S,T,V]
  float* gate_out = out + (size_t)B_ * S_ * T_ * V_;            // [B,S,G]

  // Workspace carve-out (~46.6 MB of floats).
  float* ws = (float*)d_ws;
  float* x      = ws; ws += (size_t)BB_ * H_;
  float* hA     = ws; ws += (size_t)BB_ * H_;
  float* hB     = ws; ws += (size_t)BB_ * H_;
  float* gi     = ws; ws += (size_t)BB_ * 3 * H_;
  float* gh     = ws; ws += (size_t)BB_ * 3 * H_;
  float* prob   = ws; ws += (size_t)BB_ * L_;
  float* ctx    = ws; ws += (size_t)BB_ * H_;
  float* sw     = ws; ws += (size_t)BB_;
  float* logits = ws; ws += (size_t)BB_ * V_;

  const int nBH = BB_ * H_;
  init_state_kernel<<<(nBH + 255) / 256, 256, 0, stream>>>(
      enc_hidden, slot_emb, domain_idx, slot_idx, x, hA);

  // Wave counts for the two GEMM shapes (32x80 tile per wave).
  const int gruWaves = (BB_ / 32) * (3 * H_ / 80);   // 15 * 15  = 225
  const int vocWaves = (BB_ / 32) * (V_ / 80);       // 15 * 250 = 3750
  const int gruBlks  = (gruWaves * 32 + 255) / 256;
  const int vocBlks  = (vocWaves * 32 + 255) / 256;

  float* hcur = hA;
  float* hnext = hB;
  for (int t = 0; t < T_; t++) {
    wmma_gemm_nt<<<gruBlks, 256, 0, stream>>>(x,    w_ih, gi, BB_, 3 * H_, H_);
    wmma_gemm_nt<<<gruBlks, 256, 0, stream>>>(hcur, w_hh, gh, BB_, 3 * H_, H_);
    gru_fuse_kernel<<<(nBH + 255) / 256, 256, 0, stream>>>(gi, gh, b_ih, b_hh, hcur, hnext);
    attention_kernel<<<BB_, 256, 0, stream>>>(hnext, enc_out, lens, prob, ctx);
    switch_kernel<<<(BB_ * 32 + 255) / 256, 256, 0, stream>>>(hnext, ctx, x, wr, wrb, sw);
    if (t == 0)
      gate_kernel<<<(BB_ * G_ + 255) / 256, 256, 0, stream>>>(ctx, wg, wgb, gate_out);
    wmma_gemm_nt<<<vocBlks, 256, 0, stream>>>(hnext, emb, logits, BB_, V_, H_);
    vocab_softmax_write_kernel<<<BB_, 256, 0, stream>>>(logits, sw, out, t);
    scatter_copy_kernel<<<(BB_ * L_ + 255) / 256, 256, 0, stream>>>(prob, sw, story, lens, out, t);
    teacher_gather_kernel<<<(nBH + 255) / 256, 256, 0, stream>>>(emb, targets, x, t);
    float* tmp = hcur; hcur = hnext; hnext = tmp;
  }
}